// IoUPredLoss_42107859370594
// MI455X (gfx1250) — compile-verified
//
#include <hip/hip_runtime.h>
#include <stdint.h>

// IoU-prediction loss for 7-dof lidar boxes (decode -> rotated BEV IoU -> loss).
// Purely elementwise geometry: no matmul => no WMMA. CDNA5-specific path used:
// async global->LDS staging of the shared anchor tile (ASYNCcnt) per
// CDNA5 ISA, since anchors are reused by all B=4 batches.

#define TILE_N 64   // n-values per block; block = TILE_N * B threads
#define MAXV   8

__global__ __launch_bounds__(256)
void iou_pred_loss_kernel(const float* __restrict__ iou_preds,
                          const int*   __restrict__ one_hot,
                          const float* __restrict__ weights,
                          const float* __restrict__ anchors,
                          const float* __restrict__ box_preds,
                          const float* __restrict__ reg_targets,
                          float*       __restrict__ out,
                          int B, int N) {
  __shared__ __align__(16) float s_anc[TILE_N * 7];   // 448 floats = 1792 B
  const int tid = threadIdx.x;
  const int n0  = blockIdx.x * TILE_N;

  // ---- Stage anchor tile into LDS with CDNA5 async copy (ASYNCcnt path) ----
  if (n0 + TILE_N <= N) {
    if (tid < (TILE_N * 7) / 4) {   // 112 x b128 chunks, all 16-B aligned
      uint32_t lds = (uint32_t)(uintptr_t)(&s_anc[0]) + (uint32_t)tid * 16u;
      uint64_t src = (uint64_t)(uintptr_t)(anchors + (size_t)n0 * 7 + (size_t)tid * 4);
      asm volatile("global_load_async_to_lds_b128 %0, %1, off"
                   :: "v"(lds), "v"(src) : "memory");
    }
    asm volatile("s_wait_asynccnt 0" ::: "memory");
  } else {
    const int rem = (N - n0) * 7;   // partial tail tile (not hit for N=262144)
    for (int i = tid; i < rem; i += blockDim.x)
      s_anc[i] = anchors[(size_t)n0 * 7 + i];
  }
  __syncthreads();

  const int b  = tid >> 6;          // which batch this thread handles
  const int nl = tid & 63;          // n within tile (coalesced within b-group)
  const int n  = n0 + nl;
  if (n >= N || b >= B) return;

  // ---- Anchor BEV fields from LDS (stride 7 dwords: conflict-free) ----
  const float xa   = s_anc[nl * 7 + 0];
  const float ya   = s_anc[nl * 7 + 1];
  const float dxa  = s_anc[nl * 7 + 3];
  const float dya  = s_anc[nl * 7 + 4];
  const float ra   = s_anc[nl * 7 + 6];
  const float diag = __builtin_sqrtf(dxa * dxa + dya * dya);

  const size_t gid  = (size_t)b * (size_t)N + (size_t)n;
  const size_t base = gid * 7;

  // ---- Decode the two BEV boxes (z/dz never reach the output: skip) ----
  float bx[2], by[2], bdx[2], bdy[2], br[2];
  const float* __restrict__ encs[2] = { box_preds, reg_targets };
#pragma unroll
  for (int k = 0; k < 2; ++k) {
    const float* e = encs[k] + base;
    bx[k]  = e[0] * diag + xa;
    by[k]  = e[1] * diag + ya;
    bdx[k] = __expf(e[3]) * dxa;
    bdy[k] = __expf(e[4]) * dya;
    br[k]  = e[6] + ra;
  }

  // ---- BEV corners (order matches reference: [+,-,-,+]x, [+,+,-,-]y) ----
  float c1x[4], c1y[4], c2x[4], c2y[4];
  {
    float sn, cs;
    __sincosf(br[0], &sn, &cs);
    const float hx = 0.5f * bdx[0], hy = 0.5f * bdy[0];
    c1x[0] =  hx * cs - hy * sn + bx[0];  c1y[0] =  hx * sn + hy * cs + by[0];
    c1x[1] = -hx * cs - hy * sn + bx[0];  c1y[1] = -hx * sn + hy * cs + by[0];
    c1x[2] = -hx * cs + hy * sn + bx[0];  c1y[2] = -hx * sn - hy * cs + by[0];
    c1x[3] =  hx * cs + hy * sn + bx[0];  c1y[3] =  hx * sn - hy * cs + by[0];
    __sincosf(br[1], &sn, &cs);
    const float gx = 0.5f * bdx[1], gy = 0.5f * bdy[1];
    c2x[0] =  gx * cs - gy * sn + bx[1];  c2y[0] =  gx * sn + gy * cs + by[1];
    c2x[1] = -gx * cs - gy * sn + bx[1];  c2y[1] = -gx * sn + gy * cs + by[1];
    c2x[2] = -gx * cs + gy * sn + bx[1];  c2y[2] = -gx * sn - gy * cs + by[1];
    c2x[3] =  gx * cs + gy * sn + bx[1];  c2y[3] =  gx * sn - gy * cs + by[1];
  }

  // ---- Sutherland–Hodgman: clip quad1 by the 4 half-planes of quad2 ----
  float pxv[MAXV], pyv[MAXV];
#pragma unroll
  for (int i = 0; i < MAXV; ++i) {
    pxv[i] = (i < 4) ? c1x[i] : 0.0f;
    pyv[i] = (i < 4) ? c1y[i] : 0.0f;
  }
  int nv = 4;

#pragma unroll
  for (int e = 0; e < 4; ++e) {
    const float Ax = c2x[e],            Ay = c2y[e];
    const float Bx = c2x[(e + 1) & 3],  By = c2y[(e + 1) & 3];
    const float ex = Bx - Ax,           ey = By - Ay;

    float qx[MAXV], qy[MAXV];
#pragma unroll
    for (int i = 0; i < MAXV; ++i) { qx[i] = 0.0f; qy[i] = 0.0f; }
    int m = 0;

    const int last = (nv > 0) ? nv - 1 : 0;       // single dynamic extract
    float ppx = pxv[last], ppy = pyv[last];
    float sp  = ex * (ppy - Ay) - ey * (ppx - Ax);

#pragma unroll
    for (int i = 0; i < MAXV; ++i) {
      if (i < nv) {
        const float cxv = pxv[i], cyv = pyv[i];
        const float sc  = ex * (cyv - Ay) - ey * (cxv - Ax);
        const bool cin  = (sc >= 0.0f);
        const bool pin  = (sp >= 0.0f);
        if (cin != pin) {                          // edge crosses half-plane
          float den = sp - sc;
          den = (den == 0.0f) ? 1.0f : den;        // reference's div0 guard
          const float t = sp / den;
          if (m < MAXV) { qx[m] = ppx + t * (cxv - ppx);
                          qy[m] = ppy + t * (cyv - ppy); ++m; }
        }
        if (cin && m < MAXV) { qx[m] = cxv; qy[m] = cyv; ++m; }
        ppx = cxv; ppy = cyv; sp = sc;
      }
    }
#pragma unroll
    for (int i = 0; i < MAXV; ++i) { pxv[i] = qx[i]; pyv[i] = qy[i]; }
    nv = m;
  }

  // ---- Shoelace area of the intersection polygon ----
  float ssum = 0.0f;
#pragma unroll
  for (int i = 0; i < MAXV; ++i) {
    if (i < nv) {
      const float xn = (i + 1 == nv) ? pxv[0] : pxv[(i + 1) & (MAXV - 1)];
      const float yn = (i + 1 == nv) ? pyv[0] : pyv[(i + 1) & (MAXV - 1)];
      ssum += pxv[i] * yn - xn * pyv[i];
    }
  }
  const float inter = 0.5f * __builtin_fabsf(ssum);
  const float a1    = bdx[0] * bdy[0];
  const float a2    = bdx[1] * bdy[1];
  const float iou   = inter / fmaxf(a1 + a2 - inter, 1e-6f);

  // ---- Loss: z = sigmoid(pred); p_t = log(1+exp(z)) - z*target ----
  const float tgt = (one_hot[gid] > 0) ? iou : 0.0f;
  const float z   = 1.0f / (1.0f + __expf(-iou_preds[gid]));
  const float p_t = __logf(1.0f + __expf(z)) - z * tgt;
  out[gid] = p_t * weights[gid];
}

extern "C" void kernel_launch(void* const* d_in, const int* in_sizes, int n_in,
                              void* d_out, int out_size, void* d_ws, size_t ws_size,
                              hipStream_t stream) {
  const float* iou_preds   = (const float*)d_in[0];
  const int*   one_hot     = (const int*)  d_in[1];
  const float* weights     = (const float*)d_in[2];
  const float* anchors     = (const float*)d_in[3];
  const float* box_preds   = (const float*)d_in[4];
  const float* reg_targets = (const float*)d_in[5];
  float*       out         = (float*)d_out;

  const int N = in_sizes[3] / 7;        // anchors: (N,7)
  const int B = in_sizes[0] / N;        // iou_preds: (B,N); B=4 here

  const int blocks  = (N + TILE_N - 1) / TILE_N;   // 4096 blocks
  const int threads = TILE_N * B;                  // 256 threads = 8 waves

  iou_pred_loss_kernel<<<blocks, threads, 0, stream>>>(
      iou_preds, one_hot, weights, anchors, box_preds, reg_targets, out, B, N);
}